// MGEVelocityIntr_23373212025150
// MI455X (gfx1250) — compile-verified
//
#include <hip/hip_runtime.h>
#include <math.h>

#define QUAD   128
#define NCOMP  16
#define NGRID  4096
#define SOFTR  0.01f
#define GCONST 0.004301f
#define NPIX   (4096u*4096u)
#define PI_F   3.14159265358979323846f

typedef float v2f  __attribute__((ext_vector_type(2)));
typedef float v8f  __attribute__((ext_vector_type(8)));
typedef unsigned int u32x4 __attribute__((ext_vector_type(4)));
typedef int   i32x4 __attribute__((ext_vector_type(4)));
typedef int   i32x8 __attribute__((ext_vector_type(8)));

// ---- workspace layout (float indices) ----
enum {
  WS_MAXBITS = 0,
  WS_SCALE, WS_SOFTSC, WS_BH, WS_SMUL, WS_SSUB, WS_RMIN, WS_RMAX, WS_VCOEF, WS_D, WS_L0,
  WS_A    = 64,                       // a[j*16+k], 2048 floats
  WS_C    = WS_A + QUAD*NCOMP,        // c[j*16+k], 2048 floats
  WS_GRID = WS_C + QUAD*NCOMP,        // R_grid, 4096 floats
  WS_V    = WS_GRID + NGRID,          // v_grid, 4096 floats (contiguous after R_grid)
  WS_END  = WS_V + NGRID              // ~48.3 KB total
};

#if defined(__has_builtin)
#  if __has_builtin(__builtin_amdgcn_tensor_load_to_lds)
#    define HAVE_TDM 1
#  endif
#endif

// 1-D contiguous DMA: global f32 array -> LDS, via the Tensor Data Mover.
// Call from one wave; waits TENSORcnt==0 before returning.
__device__ __forceinline__ void tdm_load_f32_1d(void* lds_dst, const float* gsrc,
                                                unsigned n_elems) {
#ifdef HAVE_TDM
  // Flat LDS pointers carry the wave-relative LDS byte offset in their low 32 bits.
  unsigned lds_addr = (unsigned)(unsigned long long)lds_dst;
  unsigned long long ga = (unsigned long long)gsrc;
  u32x4 g0;
  g0[0] = 1u;                                    // count=1, user descriptor
  g0[1] = lds_addr;                              // D#.lds_addr
  g0[2] = (unsigned)(ga & 0xFFFFFFFFu);          // global_addr[31:0]
  g0[3] = ((unsigned)(ga >> 32) & 0x01FFFFFFu)   // global_addr[56:32]
          | 0x80000000u;                         // type=2 ("image")
  i32x8 g1;
  g1[0] = (int)(2u << 16);                       // workgroup_mask=0, data_size=4B
  g1[1] = (int)((n_elems & 0xFFFFu) << 16);      // tensor_dim0[15:0]
  g1[2] = (int)((n_elems >> 16) | (1u << 16));   // tensor_dim0[31:16] | tensor_dim1=1
  g1[3] = (int)((n_elems & 0xFFFFu) << 16);      // tile_dim0 = n (n <= 65535)
  g1[4] = 1;                                     // tile_dim1=1, tile_dim2=0
  g1[5] = (int)n_elems;                          // tensor_dim0_stride lo
  g1[6] = 0;
  g1[7] = 0;
  i32x4 z4 = {0, 0, 0, 0};                       // groups 2/3 unused (<=2D tensor)
  i32x8 z8 = {0, 0, 0, 0, 0, 0, 0, 0};           // extra group (clang-23 6-arg form)
  __builtin_amdgcn_tensor_load_to_lds(g0, g1, z4, z4, z8, 0);
  __builtin_amdgcn_s_wait_tensorcnt(0);
#else
  for (unsigned i = threadIdx.x & 31u; i < n_elems; i += 32u)
    ((float*)lds_dst)[i] = gsrc[i];
#endif
}

// ---------------- Pass 1: Rmax = max(R_map) ----------------
__global__ void __launch_bounds__(256) k_max(const float* __restrict__ R,
                                             unsigned* __restrict__ maxbits) {
  const float4* R4 = (const float4*)R;
  const unsigned n4 = NPIX / 4u;
  float m = 0.0f;
  for (unsigned i = blockIdx.x * blockDim.x + threadIdx.x; i < n4;
       i += gridDim.x * blockDim.x) {
    float4 v = R4[i];
    m = fmaxf(m, fmaxf(fmaxf(v.x, v.y), fmaxf(v.z, v.w)));
  }
  // wave32 butterfly reduce
  for (int off = 16; off > 0; off >>= 1)
    m = fmaxf(m, __shfl_xor(m, off, 32));
  if ((threadIdx.x & 31u) == 0u)
    atomicMax(maxbits, __float_as_uint(m));   // all values > 0 -> uint order == float order
}

// ---------------- Pass 2: scalars, Gauss-Legendre, coefficient tables, R_grid --------
__global__ void __launch_bounds__(128) k_setup(const float* __restrict__ surf,
                                               const float* __restrict__ sigma,
                                               const float* __restrict__ qintr,
                                               const float* __restrict__ pM2L,
                                               const float* __restrict__ pInc,
                                               const float* __restrict__ pMbh,
                                               float* __restrict__ ws) {
  __shared__ double xg[QUAD], wg[QUAD];
  __shared__ float s_md[NCOMP], s_ssc[NCOMP], s_q[NCOMP];
  __shared__ float s_half, s_mid, s_d, s_L0;
  const int tid = threadIdx.x;

  // 128-pt Gauss-Legendre nodes/weights via Newton on P_128 (double).
  if (tid < QUAD / 2) {
    const int n = QUAD;
    double x = cos(3.14159265358979323846 * (tid + 0.75) / (n + 0.5));
    for (int it = 0; it < 8; ++it) {
      double p0 = 1.0, p1 = x;
      for (int l = 2; l <= n; ++l) {
        double pl = ((2.0 * l - 1.0) * x * p1 - (l - 1.0) * p0) / l;
        p0 = p1; p1 = pl;
      }
      double dp = n * (x * p1 - p0) / (x * x - 1.0);
      x -= p1 / dp;
    }
    double p0 = 1.0, p1 = x;
    for (int l = 2; l <= n; ++l) {
      double pl = ((2.0 * l - 1.0) * x * p1 - (l - 1.0) * p0) / l;
      p0 = p1; p1 = pl;
    }
    double dp = n * (x * p1 - p0) / (x * x - 1.0);
    double w = 2.0 / ((1.0 - x * x) * dp * dp);
    xg[2 * tid] = x;      wg[2 * tid] = w;
    xg[2 * tid + 1] = -x; wg[2 * tid + 1] = w;   // order irrelevant: sum over j
  }
  __syncthreads();

  if (tid == 0) {
    const float M2L = pM2L[0], inc = pInc[0], mbh = pMbh[0];
    const float si = sinf(inc), ci = cosf(inc);
    float sorted[NCOMP];
    for (int k = 0; k < NCOMP; ++k) sorted[k] = sigma[k];
    for (int i = 1; i < NCOMP; ++i) {        // insertion sort (16 elems)
      float v = sorted[i]; int j = i - 1;
      while (j >= 0 && sorted[j] > v) { sorted[j + 1] = sorted[j]; --j; }
      sorted[j + 1] = v;
    }
    const float scale = 0.5f * (sorted[7] + sorted[8]);   // quantile(., 0.5), n=16
    const float mds = 0.5f * (sorted[7] / scale + sorted[8] / scale);
    const float mxs = sorted[NCOMP - 1] / scale;
    const float lo = asinhf(logf(1e-7f * mds) * (2.0f / PI_F));
    const float hi = asinhf(logf(1000.0f * mxs) * (2.0f / PI_F));
    s_half = 0.5f * (hi - lo);
    s_mid  = 0.5f * (hi + lo);
    const float SQ2PI = 2.50662827463100050f;
    for (int k = 0; k < NCOMP; ++k) {
      const float q = qintr[k];
      const float qobs = sqrtf(q * q * si * si + ci * ci);
      s_md[k]  = surf[k] * M2L * qobs / (q * sigma[k] * SQ2PI);
      s_ssc[k] = sigma[k] / scale;
      s_q[k]   = q;
    }
    const float Rmax = __uint_as_float(((const unsigned*)ws)[WS_MAXBITS]);
    const float Rmin = SOFTR;
    const float L0 = log10f(Rmin);
    const float d  = (log10f(Rmax) - L0) / (float)(NGRID - 1);
    s_d = d; s_L0 = L0;
    ws[WS_SCALE]  = scale;
    ws[WS_SOFTSC] = SOFTR / scale;
    ws[WS_BH]     = GCONST * exp10f(mbh) / scale;
    ws[WS_SMUL]   = 0.30102999566398120f / d;   // log10(2)/d
    ws[WS_SSUB]   = L0 / d;
    ws[WS_RMIN]   = Rmin;
    ws[WS_RMAX]   = Rmax;
    ws[WS_VCOEF]  = 2.0f * PI_F * GCONST * scale * scale;
    ws[WS_D]      = d;
    ws[WS_L0]     = L0;
  }
  __syncthreads();

  // per-(j,k) coefficients: term(i,j,k) = c[j,k] * exp(a[j,k] * R_sc_i^2)
  {
    const float t  = s_half * (float)xg[tid] + s_mid;
    const float u  = expf(0.5f * PI_F * sinhf(t));
    const float du = 0.5f * PI_F * coshf(t) * u;
    const float wj = s_half * (float)wg[tid];
    const float onep = 1.0f + u;
    const float wdu = wj * du / (onep * onep);
    for (int k = 0; k < NCOMP; ++k) {
      const int idx = tid * NCOMP + k;
      ws[WS_A + idx] = -0.5f / (s_ssc[k] * s_ssc[k] * onep);
      ws[WS_C + idx] = s_q[k] * s_md[k] * wdu * rsqrtf(s_q[k] * s_q[k] + u);
    }
  }

  // log-spaced radius grid
  for (int i = tid; i < NGRID; i += blockDim.x)
    ws[WS_GRID + i] = exp10f(s_L0 + (float)i * s_d);
}

// ---------------- Pass 3: v_grid via WMMA GEMV over quadrature terms ----------------
// One wave owns 16 radii (M). K=2048 terms consumed 4 at a time with
// v_wmma_f32_16x16x4_f32; B broadcasts c[k] across N, so every D column = y.
__global__ void __launch_bounds__(128) k_vgrid(float* __restrict__ ws) {
  __shared__ float ac[2 * QUAD * NCOMP];     // a[2048] then c[2048], 16 KB
  if (threadIdx.x < 32u) tdm_load_f32_1d(ac, ws + WS_A, 2 * QUAD * NCOMP);
  __syncthreads();
  const float* At = ac;
  const float* Ct = ac + QUAD * NCOMP;

  const unsigned wave = (blockIdx.x * blockDim.x + threadIdx.x) >> 5;
  const unsigned lane = threadIdx.x & 31u;
  const unsigned base = wave * 16u;

  const float inv_scale = 1.0f / ws[WS_SCALE];
  const float r_sc = ws[WS_GRID + base + (lane & 15u)] * inv_scale;
  const float r2 = r_sc * r_sc;
  // A 16x4 f32 layout: lanes 0-15 hold K={0,1}, lanes 16-31 hold K={2,3}; B matches.
  const int ksel = (int)((lane & 16u) >> 3);   // 0 or 2

  v8f acc = {};
  for (int kb = 0; kb < QUAD * NCOMP; kb += 4) {
    const int k0 = kb + ksel;
    v2f Am, Bm;
    Am.x = __expf(At[k0] * r2);
    Am.y = __expf(At[k0 + 1] * r2);
    Bm.x = Ct[k0];
    Bm.y = Ct[k0 + 1];
    acc = __builtin_amdgcn_wmma_f32_16x16x4_f32(false, Am, false, Bm,
                                                (short)0, acc, false, false);
  }

  // D layout: lane L<16 VGPR r -> (M=r, N=L); lane L>=16 -> (M=8+r). Columns identical.
  if ((lane & 15u) == 0u) {
    const unsigned mbase = base + ((lane >> 4) << 3);
    const float vcoef = ws[WS_VCOEF], bh = ws[WS_BH], soft = ws[WS_SOFTSC];
#pragma unroll
    for (int r = 0; r < 8; ++r) {
      const float rs = ws[WS_GRID + mbase + r] * inv_scale;
      const float x = rs * rs + soft * soft;
      const float rr = rsqrtf(x);                 // x^-1.5 = rr^3
      const float vc2 = vcoef * acc[r] + bh * (rr * rr * rr);
      ws[WS_V + mbase + r] = rs * sqrtf(vc2);
    }
  }
}

// ---------------- Pass 4: streaming interpolation (bandwidth-bound) ----------------
__global__ void __launch_bounds__(256) k_interp(const float* __restrict__ R,
                                                float* __restrict__ out,
                                                const float* __restrict__ ws) {
  __shared__ float tab[2 * NGRID];   // R_grid | v_grid, 32 KB, DMA'd once per block
  if (threadIdx.x < 32u) tdm_load_f32_1d(tab, ws + WS_GRID, 2 * NGRID);
  __syncthreads();
  const float* grid = tab;
  const float* vg   = tab + NGRID;

  const float Rmin = ws[WS_RMIN], Rmax = ws[WS_RMAX];
  const float smul = ws[WS_SMUL], ssub = ws[WS_SSUB];

  const float4* R4 = (const float4*)R;
  float4* O4 = (float4*)out;
  const unsigned n4 = NPIX / 4u;
  for (unsigned i = blockIdx.x * blockDim.x + threadIdx.x; i < n4;
       i += gridDim.x * blockDim.x) {
    float4 rv = R4[i];
    float4 ov;
    float* rp = &rv.x;
    float* op = &ov.x;
#pragma unroll
    for (int c = 0; c < 4; ++c) {
      const float Rc = fminf(fmaxf(rp[c], Rmin), Rmax);
      // analytic searchsorted on a log grid, then exact fixup vs stored grid
      const float tf = __log2f(Rc) * smul - ssub;
      int h = (int)ceilf(tf);
      h = min(max(h, 0), NGRID - 1);
      while (h > 0 && grid[h - 1] >= Rc) --h;
      while (h < NGRID - 1 && grid[h] < Rc) ++h;
      const int lo2 = (h > 0) ? h - 1 : 0;
      const float Rl = grid[lo2], Rh = grid[h];
      const float w = (Rh == Rl) ? 0.0f : (Rc - Rl) / (Rh - Rl);
      op[c] = vg[lo2] + w * (vg[h] - vg[lo2]);
    }
    O4[i] = ov;
  }
}

extern "C" void kernel_launch(void* const* d_in, const int* in_sizes, int n_in,
                              void* d_out, int out_size, void* d_ws, size_t ws_size,
                              hipStream_t stream) {
  (void)in_sizes; (void)n_in; (void)out_size; (void)ws_size;
  const float* R    = (const float*)d_in[0];
  const float* surf = (const float*)d_in[1];
  const float* sig  = (const float*)d_in[2];
  const float* q    = (const float*)d_in[3];
  const float* m2l  = (const float*)d_in[4];
  const float* inc  = (const float*)d_in[5];
  const float* mbh  = (const float*)d_in[6];
  float* out = (float*)d_out;
  float* ws  = (float*)d_ws;

  (void)hipMemsetAsync(d_ws, 0, sizeof(unsigned), stream);     // max accumulator = 0
  k_max<<<2048, 256, 0, stream>>>(R, (unsigned*)d_ws);         // 64 MB sweep -> L2 resident
  k_setup<<<1, 128, 0, stream>>>(surf, sig, q, m2l, inc, mbh, ws);
  k_vgrid<<<64, 128, 0, stream>>>(ws);                         // 256 waves x 16 radii
  k_interp<<<2048, 256, 0, stream>>>(R, out, ws);              // streaming pass
}